// Convolution_90340342104442
// MI455X (gfx1250) — compile-verified
//
#include <hip/hip_runtime.h>
#include <stdint.h>

// ---------------- problem constants (match reference) ----------------
#define BB 4
#define CC 32
#define HH 32
#define WW 32
#define OO 32
#define KK 4
#define LL (HH * WW)
#define NEIGH 8
#define VSAMP 12
#define EPSF 1e-6f

typedef __attribute__((ext_vector_type(2))) float v2f;
typedef __attribute__((ext_vector_type(8))) float v8f;
typedef int v4i __attribute__((__vector_size__(4 * sizeof(int))));  // matches builtin proto

#if defined(__AMDGCN__) && __has_builtin(__builtin_amdgcn_wmma_f32_16x16x4_f32)
#define HAVE_WMMA44 1
#else
#define HAVE_WMMA44 0
#endif

#if defined(__AMDGCN__) && __has_builtin(__builtin_amdgcn_global_load_async_to_lds_b128)
#define HAVE_ASYNC_LD 1
#else
#define HAVE_ASYNC_LD 0
#endif

#if defined(__AMDGCN__) && __has_builtin(__builtin_amdgcn_global_store_async_from_lds_b128)
#define HAVE_ASYNC_ST 1
#else
#define HAVE_ASYNC_ST 0
#endif

#if defined(__AMDGCN__)
// typed, addrspace-qualified casts for the async-LDS builtins
// (param type per compiler diagnostic: 'int __vector(4) __device__ *')
#define ASG(p) ((__attribute__((address_space(1))) v4i*)(p))
#define ASL(p) ((__attribute__((address_space(3))) v4i*)(p))
__device__ __forceinline__ void wait_async_all() {
  asm volatile("s_wait_asynccnt 0" ::: "memory");
}
#else
__device__ __forceinline__ void wait_async_all() {}
#endif

// Deterministic counter-based PRNG (stateless, graph-capture safe).
__device__ __forceinline__ uint32_t pcg_hash(uint32_t v) {
  uint32_t s = v * 747796405u + 2891336453u;
  uint32_t w = ((s >> ((s >> 28u) + 4u)) ^ s) * 277803737u;
  return (w >> 22u) ^ w;
}
__device__ __forceinline__ float u01(uint32_t v) {
  // uniform in [0,1), then * (1-EPS) like the reference
  return (float)(pcg_hash(v) >> 8) * (1.0f / 16777216.0f) * (1.0f - EPSF);
}

// One workgroup = (batch b, image row y). 8 waves x 4 pixels each = 32 pixels.
// Each wave handles one pixel l at a time, producing all O=32 channels.
__global__ __launch_bounds__(256) void adaptive_conv_fused(
    const float* __restrict__ x, const float* __restrict__ means_raw,
    const float* __restrict__ sigmas_raw, const float* __restrict__ values,
    const float* __restrict__ bias, float* __restrict__ out) {
  __shared__ float lds_x[3 * CC * WW];      // [dy][c][x]: rows y-1..y+1, OOB rows zero
  __shared__ float lds_mean[OO * KK * 3];   // transformed means
  __shared__ float lds_isig[OO * KK * 3];   // 1/sqrt(eps+sigma)
  __shared__ float lds_low[OO * KK * 3];    // local-sample lower bounds
  __shared__ float lds_val[OO * KK];        // values[o,k]
  __shared__ float lds_out[OO * WW];        // [o][x] results staged for writeback

  const int tid = threadIdx.x;
  const int lane = tid & 31;
  const int wave = tid >> 5;
  const int b = blockIdx.x / HH;
  const int y = blockIdx.x % HH;

  // gfx1250 prefetch hint for this block's x slab (global_prefetch_b8)
  __builtin_prefetch(x + (size_t)(b * CC) * HH * WW + (size_t)y * WW, 0, 1);

  // ---- per-(o,k) parameter transforms into LDS ----
  if (tid < OO * KK) {
    const float sr = sigmas_raw[tid];
    const float s = log1pf(expf(sr + 2.0f)) + EPSF;  // softplus(raw+SIGMA_BOOST)+EPS
#pragma unroll
    for (int d = 0; d < 3; ++d) {
      const float SZ = (d == 0) ? 32.0f : 3.0f;  // SIZE == RNG
      const float RG = (d == 0) ? 6.0f : 2.0f;   // REGION = (floor(0.2*C), KS-1, KS-1)
      const float raw = means_raw[tid * 3 + d];
      const float m = (SZ - 1.0f) / (1.0f + expf(-raw));  // sigmoid(raw)*(SIZE-1)
      const float sig = s * SZ * 0.1f;                    // SIGMA_SCALE, MIN_SIGMA=0
      const float isg = sqrtf(1.0f / (EPSF + sig));
      const float mr = rintf(m);  // jnp.round = round-half-even
      float lo = mr - RG * 0.5f;
      const float up = mr + RG * 0.5f;
      lo = fmaxf(lo, 0.0f);
      if (up > SZ) lo = SZ - RG;
      lds_mean[tid * 3 + d] = m;
      lds_isig[tid * 3 + d] = isg;
      lds_low[tid * 3 + d] = lo;
    }
    lds_val[tid] = values[tid];
  }

  // ---- stage x rows y-1..y+1 into LDS (zero padded) ----
#if HAVE_ASYNC_LD
  {
    // 768 b128 chunks; iteration index == dy, so row validity is wave-uniform.
    const float4 z4 = {0.0f, 0.0f, 0.0f, 0.0f};
#pragma unroll
    for (int it = 0; it < 3; ++it) {
      const int cidx = it * 256 + tid;   // 16-byte chunk id in lds_x
      const int c = (cidx >> 3) & 31;    // channel
      const int xg = (cidx & 7) << 2;    // x-group start (4 floats)
      const int yy = y - 1 + it;
      if (yy >= 0 && yy < HH) {
        // ASYNCcnt-tracked DMA: global -> LDS, no VGPR round trip
        __builtin_amdgcn_global_load_async_to_lds_b128(
            ASG(x + (size_t)((b * CC + c) * HH + yy) * WW + xg),
            ASL(&lds_x[cidx << 2]), 0, 0);
      } else {
        *(float4*)&lds_x[cidx << 2] = z4;  // zero-pad OOB image rows
      }
    }
  }
  wait_async_all();  // s_wait_asynccnt 0: our async loads landed in LDS
#else
  for (int i = tid; i < 3 * CC * WW; i += 256) {
    const int dy = i >> 10;
    const int c = (i >> 5) & 31;
    const int xx = i & 31;
    const int yy = y - 1 + dy;
    float v = 0.0f;
    if (yy >= 0 && yy < HH) v = x[(size_t)((b * CC + c) * HH + yy) * WW + xx];
    lds_x[i] = v;
  }
  wait_async_all();
#endif

#if defined(__AMDGCN__)
#if __has_builtin(__builtin_amdgcn_s_wait_tensorcnt)
  __builtin_amdgcn_s_wait_tensorcnt(0);  // CDNA5 split counters (no TDM in flight: no-op)
#endif
#if __has_builtin(__builtin_amdgcn_s_cluster_barrier)
  // Wave-cluster barrier path (s_barrier_signal/-3 + s_barrier_wait/-3).
  // Runtime NOP when the grid is not dispatched as clusters (ClusterID==0).
  __builtin_amdgcn_s_cluster_barrier();
#endif
#endif
  __syncthreads();

  // WMMA A-matrix layout (f32 16x4): lanes 0-15 row m hold K=0,1; lanes 16-31 hold K=2,3.
  const int m16 = lane & 15;
  const int half = lane >> 4;
  const int k0 = 2 * half, k1 = 2 * half + 1;

  for (int xi = 0; xi < 4; ++xi) {
    const int xpos = wave * 4 + xi;
    const int l = y * WW + xpos;
    float acc = 0.0f;  // this lane keeps the result for o == lane

    for (int o = 0; o < OO; ++o) {
      float ea[3], eb[3], pv[3];
      float s0 = 0.0f, s1 = 0.0f;
#pragma unroll
      for (int blk = 0; blk < 3; ++blk) {
        const int i = blk * 16 + m16;  // sample index 0..47
        const int ki = i / VSAMP;      // gaussian that generated this sample
        const int vs = i - ki * VSAMP;
        const int pbase = (o * KK + ki) * 3;
        float p0, p1, p2;
        if (vs < NEIGH) {
          // floor/ceil neighbor combos; FM row vs: bit d true (->floor) if ((vs>>(2-d))&1)==0
          p0 = (((vs >> 2) & 1) == 0) ? floorf(lds_mean[pbase + 0]) : ceilf(lds_mean[pbase + 0]);
          p1 = (((vs >> 1) & 1) == 0) ? floorf(lds_mean[pbase + 1]) : ceilf(lds_mean[pbase + 1]);
          p2 = (((vs >> 0) & 1) == 0) ? floorf(lds_mean[pbase + 2]) : ceilf(lds_mean[pbase + 2]);
        } else if (vs < NEIGH + 2) {
          // global uniform samples: floor(u * RNG)
          const uint32_t a = (uint32_t)(vs - NEIGH);
          const uint32_t seed =
              ((((uint32_t)(b * LL + l) * OO + (uint32_t)o) * KK + (uint32_t)ki) * 2u + a) * 3u;
          p0 = floorf(u01(seed + 0u) * 32.0f);
          p1 = floorf(u01(seed + 1u) * 3.0f);
          p2 = floorf(u01(seed + 2u) * 3.0f);
        } else {
          // local samples: floor(u * REGION + lower)
          const uint32_t a = (uint32_t)(vs - NEIGH - 2);
          const uint32_t seed =
              (((((uint32_t)(b * LL + l) * OO + (uint32_t)o) * KK + (uint32_t)ki) * 2u + a) * 3u) ^
              0x9e3779b9u;
          p0 = floorf(u01(seed + 0u) * 6.0f + lds_low[pbase + 0]);
          p1 = floorf(u01(seed + 1u) * 2.0f + lds_low[pbase + 1]);
          p2 = floorf(u01(seed + 2u) * 2.0f + lds_low[pbase + 2]);
        }
        // gather pflat[b, l, c*9 + dy*3 + dx] = xpad[b, c, y+dy, x+dx]
        const int gc = (int)p0, gdy = (int)p1, gdx = (int)p2;
        const int xc = xpos + gdx - 1;
        float patch = 0.0f;
        if (xc >= 0 && xc < WW) patch = lds_x[gdy * (CC * WW) + gc * WW + xc];
        pv[blk] = patch;
        // unnormalized gaussian responsibilities for this lane's two k's
        const int q0 = (o * KK + k0) * 3, q1 = (o * KK + k1) * 3;
        float d0 = (p0 - lds_mean[q0 + 0]) * lds_isig[q0 + 0];
        float d1 = (p1 - lds_mean[q0 + 1]) * lds_isig[q0 + 1];
        float d2 = (p2 - lds_mean[q0 + 2]) * lds_isig[q0 + 2];
        ea[blk] = expf(-0.5f * (d0 * d0 + d1 * d1 + d2 * d2));
        d0 = (p0 - lds_mean[q1 + 0]) * lds_isig[q1 + 0];
        d1 = (p1 - lds_mean[q1 + 1]) * lds_isig[q1 + 1];
        d2 = (p2 - lds_mean[q1 + 2]) * lds_isig[q1 + 2];
        eb[blk] = expf(-0.5f * (d0 * d0 + d1 * d1 + d2 * d2));
        s0 += ea[blk];
        s1 += eb[blk];
      }
      // S[k] = sum_i e[i,k]: reduce across the 16 lanes of this half (xor<16 stays in half)
#pragma unroll
      for (int off = 1; off < 16; off <<= 1) {
        s0 += __shfl_xor(s0, off, 32);
        s1 += __shfl_xor(s1, off, 32);
      }
      const float vb0 = lds_val[o * KK + k0] / s0;  // values[o,k]/S[k]
      const float vb1 = lds_val[o * KK + k1] / s1;
      float resv;
#if HAVE_WMMA44
      // out_o = sum_i sum_k (e[i,k]*patch[i]) * (values[o,k]/S[k])
      // A (16x4 f32) rows = e*patch, B (4x16) columns all = V', chain C over 3 row blocks,
      // then column-sum of D. v_wmma_f32_16x16x4_f32: K=4 matches the einsum exactly.
      v2f bop;
      bop.x = vb0;
      bop.y = vb1;
      v8f c8 = {0.0f, 0.0f, 0.0f, 0.0f, 0.0f, 0.0f, 0.0f, 0.0f};
#pragma unroll
      for (int blk = 0; blk < 3; ++blk) {
        v2f aop;
        aop.x = ea[blk] * pv[blk];
        aop.y = eb[blk] * pv[blk];
        c8 = __builtin_amdgcn_wmma_f32_16x16x4_f32(false, aop, false, bop, (short)0, c8,
                                                   false, false);
      }
      float partial = ((c8[0] + c8[1]) + (c8[2] + c8[3])) + ((c8[4] + c8[5]) + (c8[6] + c8[7]));
      resv = partial + __shfl_xor(partial, 16, 32);  // add the other half's 8 rows
#else
      float partial = 0.0f;
#pragma unroll
      for (int blk = 0; blk < 3; ++blk) partial += pv[blk] * (ea[blk] * vb0 + eb[blk] * vb1);
#pragma unroll
      for (int off = 1; off < 32; off <<= 1) partial += __shfl_xor(partial, off, 32);
      resv = partial;
#endif
      if (lane == o) acc = resv;
    }
    // stage result (+bias) in LDS for the async writeback: lds_out[o][x]
    lds_out[lane * WW + xpos] = acc + bias[(size_t)(lane * HH + y) * WW + xpos];
  }

  __syncthreads();  // all 1024 results of this (b,y) row-slab are in lds_out

  // ---- write back out[b,:,y,:] ----
#if HAVE_ASYNC_ST
  {
    // 256 b128 chunks = 1024 floats; one chunk per thread, ASYNCcnt-tracked LDS -> global DMA
    const int o = tid >> 3;
    const int xg = (tid & 7) << 2;
    __builtin_amdgcn_global_store_async_from_lds_b128(
        ASG(out + (size_t)((b * OO + o) * HH + y) * WW + xg),
        ASL(&lds_out[o * WW + xg]), 0, 0);
  }
  wait_async_all();  // drain async stores before wave exit
#else
  for (int t = tid; t < OO * WW; t += 256) {
    const int o = t >> 5;
    const int xx = t & 31;
    out[(size_t)((b * OO + o) * HH + y) * WW + xx] = lds_out[t];
  }
#endif
}

extern "C" void kernel_launch(void* const* d_in, const int* in_sizes, int n_in, void* d_out,
                              int out_size, void* d_ws, size_t ws_size, hipStream_t stream) {
  (void)in_sizes;
  (void)n_in;
  (void)out_size;
  (void)d_ws;
  (void)ws_size;
  const float* x = (const float*)d_in[0];
  const float* means_raw = (const float*)d_in[1];
  const float* sigmas_raw = (const float*)d_in[2];
  const float* values = (const float*)d_in[3];
  const float* bias = (const float*)d_in[4];
  float* out = (float*)d_out;

  dim3 grid(BB * HH);  // one workgroup per (batch, image row)
  dim3 block(256);     // 8 wave32 waves
  hipLaunchKernelGGL(adaptive_conv_fused, grid, block, 0, stream, x, means_raw, sigmas_raw,
                     values, bias, out);
}